// FlowmatchingActionHead_54631984005774
// MI455X (gfx1250) — compile-verified
//
#include <hip/hip_runtime.h>
#include <hip/hip_bf16.h>
#include <stdint.h>

// ---------------------------------------------------------------------------
// FlowmatchingActionHead on gfx1250 (MI455X).
//
// B=32, T=16, A=32, D=1536, E=32.
//   x  = [actions@W1[cid]+b1 | sinusoidal_pe(t)]    (per batch 16 x 3072)
//   h  = swish(x@W2[cid]+b2)                         (16 x 1536)
//   out= h@W3[cid]+b3                                (16 x 1536)
//
// HBM-bound: ~0.9GB of f32 weight streaming vs 7.3 GFLOP (~8 FLOP/B).
// -> keep f32, V_WMMA_F32_16X16X4_F32 (M=16 == one WMMA tile row).
// Activations are staged into LDS once per block via the Tensor Data Mover
// (tensor_load_to_lds + s_wait_tensorcnt), stored k-major in the workspace so
// the TDM copy is a contiguous 1D tile and LDS reads are bank-conflict-free.
// ---------------------------------------------------------------------------

typedef __attribute__((ext_vector_type(2))) float v2f;
typedef __attribute__((ext_vector_type(8))) float v8f;
typedef __attribute__((ext_vector_type(4))) unsigned int v4u;
typedef __attribute__((ext_vector_type(8))) int v8i;
typedef __attribute__((ext_vector_type(4))) int v4i;

#define NB   32      // batches
#define NT_  16      // rows (T)
#define DD   1536    // hidden dim (N of every layer)

// One f32 WMMA k-step: D(16x16) += A(16x4) * B(4x16)
__device__ __forceinline__ v8f wmma_step(v2f a, v2f b, v8f c) {
    // 8 args: (neg_a, A, neg_b, B, c_mod, C, reuse_a, reuse_b)
    return __builtin_amdgcn_wmma_f32_16x16x4_f32(
        false, a, false, b, (short)0, c, false, false);
}

// TDM: async-copy `ndwords` contiguous dwords from global `gsrc` to LDS offset
// `lds_off`. 1D tile descriptor per CDNA5 ISA:
//   group0: count=1 | lds_addr | global_addr[56:0] | type=2
//   group1: data_size=4B, tensor_dim0 = tile_dim0 = ndwords, tensor_dim1=1
// This toolchain (clang-23 / therock-10.0) uses the 6-arg builtin:
//   (uint32x4 g0, int32x8 g1, int32x4 g2, int32x4 g3, int32x8 g4, i32 cpol)
__device__ __forceinline__ void tdm_load_1d(const float* gsrc,
                                            unsigned int lds_off,
                                            unsigned int ndwords) {
    const unsigned long long ga = (unsigned long long)(uintptr_t)gsrc;
    v4u g0;
    g0[0] = 1u;                                       // count=1, user desc
    g0[1] = lds_off;                                  // lds_addr (bytes)
    g0[2] = (unsigned int)(ga & 0xffffffffu);         // global_addr[31:0]
    g0[3] = (unsigned int)((ga >> 32) & 0x01ffffffu)  // global_addr[56:32]
          | (2u << 30);                               // type = 2 ("image")
    v8i g1;
    g1[0] = (int)(2u << 16);                          // data_size = 4 bytes
    g1[1] = (int)((ndwords & 0xffffu) << 16);         // tensor_dim0[15:0]
    g1[2] = (int)(((ndwords >> 16) & 0xffffu)         // tensor_dim0[31:16]
          | (1u << 16));                              // tensor_dim1 = 1
    g1[3] = (int)((ndwords & 0xffffu) << 16);         // tile_dim0 = ndwords
    g1[4] = 0;                                        // tile_dim1/2 unused
    g1[5] = (int)ndwords;                             // tensor_dim0_stride
    g1[6] = 0;
    g1[7] = 0;
    v4i g2 = {0, 0, 0, 0};
    v4i g3 = {0, 0, 0, 0};
    v8i g4 = {0, 0, 0, 0, 0, 0, 0, 0};
    __builtin_amdgcn_tensor_load_to_lds(g0, g1, g2, g3, g4, 0);
}

// ---------------------------------------------------------------------------
// Per-category GEMM: out = act( in(16,K) @ W[cid](K,1536) + bias[cid] )
// K compile-time. Grid (B, 12) x 256 threads (8 waves); wave w of block y owns
// output N-tile y*8+w (96 tiles of 16 cols).
//
// STAGE=true : `in` is k-major (B, K, 16); staged chunk-wise into LDS by TDM.
// STAGE=false: `in` is row-major (B, 16, K); fragments loaded from global.
// OUT_T=true : output written k-major (B, 1536, 16) for the next layer's TDM.
// ACT: 0 identity, 1 swish.
// ---------------------------------------------------------------------------
template <int K, int KC, bool STAGE, bool OUT_T, int ACT>
__global__ __launch_bounds__(256) void cat_gemm_kernel(
    const float* __restrict__ in,
    const float* __restrict__ W,       // (E, K, 1536) row-major
    const float* __restrict__ bias,    // (E, 1536)
    const int*   __restrict__ cat_ids, // (B,)
    float*       __restrict__ out,
    int outBatchStride)
{
    __shared__ float smem[(STAGE ? KC : 1) * NT_];   // 48KB max (KC=768)

    const int b    = blockIdx.x;
    const int wave = threadIdx.x >> 5;
    const int lane = threadIdx.x & 31;
    const int nt   = blockIdx.y * 8 + wave;   // 0..95
    const int col  = nt * 16 + (lane & 15);   // output / B-matrix column
    const int row  = lane & 15;               // A-matrix row (M)
    const int kro  = (lane >> 4) << 1;        // lanes 0-15 -> K{0,1}, 16-31 -> K{2,3}

    const int cid = cat_ids[b];

    // WMMA f32 fragment maps (CDNA5 ISA):
    //  A 16x4 : lane L<16 holds A[L][k+0..1]; lane L>=16 holds A[L-16][k+2..3]
    //  B 4x16 : lane L<16 holds B[k+0..1][n+L]; lane L>=16 holds B[k+2..3][n+L-16]
    //  C 16x16: VGPR v -> (M = v + 8*(lane>=16), N = n + (lane&15))
    const float* wp = W + (size_t)cid * K * DD + (size_t)kro * DD + col;

    v8f c = {0.f, 0.f, 0.f, 0.f, 0.f, 0.f, 0.f, 0.f};

    if constexpr (STAGE) {
        // Conflict-free LDS reads: element (k,r) at smem[k*16+r]; lanes 0-15
        // hit banks X..X+15, lanes 16-31 hit X+32..X+47.
        const float* slab = in + (size_t)b * K * NT_;   // k-major batch slab
        const unsigned int smem_off = (unsigned int)(uintptr_t)&smem[0];

        for (int k0 = 0; k0 < K; k0 += KC) {
            __syncthreads();                 // previous chunk fully consumed
            if (threadIdx.x < 32) {          // wave 0 drives the TDM
                tdm_load_1d(slab + (size_t)k0 * NT_, smem_off, KC * NT_);
                __builtin_amdgcn_s_wait_tensorcnt(0);
            }
            __syncthreads();                 // chunk visible to all waves

            // Warm GL2 one weight-chunk ahead of the stream.
            if (k0 + KC < K) {
                __builtin_prefetch(wp + (size_t)KC * DD, 0, 0);
            }

#pragma unroll 8
            for (int kk = 0; kk < KC; kk += 4) {
                v2f a;
                a.x = smem[(kk + kro) * NT_ + row];
                a.y = smem[(kk + kro + 1) * NT_ + row];
                v2f bb;
                bb.x = wp[0];                // W[k+kro  ][col]
                bb.y = wp[DD];               // W[k+kro+1][col]
                c = wmma_step(a, bb, c);
                wp += (size_t)4 * DD;
            }
        }
    } else {
        // Row-major global A (tiny layer-1 slab: 16xK, L2/L0 resident).
        const float* ap = in + (size_t)b * NT_ * K + (size_t)row * K + kro;
#pragma unroll
        for (int k = 0; k < K; k += 4) {
            v2f a = *(const v2f*)ap;         // contiguous float2
            v2f bb;
            bb.x = wp[0];
            bb.y = wp[DD];
            c = wmma_step(a, bb, c);
            ap += 4;
            wp += (size_t)4 * DD;
        }
    }

    const float bv = bias[(size_t)cid * DD + col];
    float* op = out + (size_t)b * outBatchStride;
    const int rbase = (lane >> 4) << 3;       // 0 or 8
#pragma unroll
    for (int v = 0; v < 8; ++v) {
        float y = c[v] + bv;
        if (ACT == 1) {
            y = y * (1.0f / (1.0f + __expf(-y)));   // swish
        }
        if (OUT_T) {
            op[(size_t)col * NT_ + rbase + v] = y;   // k-major: 8 contiguous
        } else {
            op[(size_t)(rbase + v) * DD + col] = y;  // row-major (final out)
        }
    }
}

// Sinusoidal embedding of the per-batch scalar timestep, written k-major into
// x[b, 1536+j, r] (sin) and x[b, 2304+j, r] (cos): 16 contiguous floats each.
__global__ __launch_bounds__(256) void tau_pe_kernel(
    const int* __restrict__ timesteps,   // (B,)
    float*     __restrict__ xT)          // (B, 3072, 16) k-major
{
    const int b = blockIdx.x;
    const float t = (float)timesteps[b];
    float* xb = xT + (size_t)b * 3072 * NT_;

    for (int j = threadIdx.x; j < 768; j += blockDim.x) {
        // ln(10000) = 9.210340371976184
        const float freq = __expf(-9.210340372f * (float)j * (1.0f / 768.0f));
        const float ang  = t * freq;
        const float s  = sinf(ang);
        const float cc = cosf(ang);
        float* ps = xb + (size_t)(DD + j) * NT_;
        float* pc = xb + (size_t)(DD + 768 + j) * NT_;
#pragma unroll
        for (int r = 0; r < NT_; ++r) {
            ps[r] = s;
            pc[r] = cc;
        }
    }
}

extern "C" void kernel_launch(void* const* d_in, const int* in_sizes, int n_in,
                              void* d_out, int out_size, void* d_ws, size_t ws_size,
                              hipStream_t stream) {
    // setup_inputs() order:
    const float* actions   = (const float*)d_in[0];  // (B,16,32)
    const int*   timesteps = (const int*)  d_in[1];  // (B,)
    const int*   cat_ids   = (const int*)  d_in[2];  // (B,)
    const float* W1        = (const float*)d_in[3];  // (E,32,1536)
    const float* b1        = (const float*)d_in[4];  // (E,1536)
    const float* W2        = (const float*)d_in[5];  // (E,3072,1536)
    const float* b2        = (const float*)d_in[6];  // (E,1536)
    const float* W3        = (const float*)d_in[7];  // (E,1536,1536)
    const float* b3        = (const float*)d_in[8];  // (E,1536)
    float* outp            = (float*)d_out;          // (B,16,1536)

    // Workspace (k-major activations): xT (B,3072,16) = 6MB, hT (B,1536,16) = 3MB
    float* xT = (float*)d_ws;
    float* hT = xT + (size_t)NB * 3072 * NT_;

    const dim3 grid(NB, 12);   // 96 N-tiles / 8 waves-per-block
    const dim3 blk(256);

    // Layer 1: a_emb -> xT[:, 0:1536, :]   (row-major input, k-major output)
    cat_gemm_kernel<32, 32, false, true, 0>
        <<<grid, blk, 0, stream>>>(actions, W1, b1, cat_ids, xT, 3072 * NT_);
    // tau embedding -> xT[:, 1536:3072, :]
    tau_pe_kernel<<<NB, blk, 0, stream>>>(timesteps, xT);
    // Layer 2 (swish): hT = swish(x @ W2 + b2)   (TDM-staged A, k-major out)
    cat_gemm_kernel<3072, 768, true, true, 1>
        <<<grid, blk, 0, stream>>>(xT, W2, b2, cat_ids, hT, DD * NT_);
    // Layer 3: out = h @ W3 + b3                 (TDM-staged A, row-major out)
    cat_gemm_kernel<1536, 768, true, false, 0>
        <<<grid, blk, 0, stream>>>(hT, W3, b3, cat_ids, outp, NT_ * DD);
}